// SparseMultiDense_26414048870602
// MI455X (gfx1250) — compile-verified
//
#include <hip/hip_runtime.h>

typedef __attribute__((ext_vector_type(2))) float v2f;
typedef __attribute__((ext_vector_type(8))) float v8f;

#define TASKS          8
#define BATCH          32768
#define IN_DIM         128
#define OUT_DIM        128
#define ROWS_PER_BLOCK 128
#define K_STEPS        (IN_DIM / 4)    // 32 wmma k-steps of K=4
#define N_TILES        (OUT_DIM / 16)  // 8

// out[t] = in[t] (32768x128) @ w[t] (128x128) + b[t], f32, via V_WMMA_F32_16X16X4_F32.
// One wave -> 16 rows x full N=128 (8 accumulator tiles). Weights staged in LDS in
// WMMA B-fragment order so the inner loop is ds_load_b64 with immediate offsets.
__global__ __launch_bounds__(256)
void smd_wmma_f32_kernel(const float* __restrict__ in,
                         const float* __restrict__ w,
                         const float* __restrict__ b,
                         float* __restrict__ out)
{
    // B fragments: [ks][nt][lane] -> float2.  64 KB.
    // lane<16 : cols n=nt*16+lane, rows K=4ks, 4ks+1
    // lane>=16: cols n=nt*16+lane-16, rows K=4ks+2, 4ks+3
    __shared__ v2f wfrag[K_STEPS * N_TILES * 32];

    const int tid  = threadIdx.x;
    const int lane = tid & 31;
    const int wave = tid >> 5;
    const int task = blockIdx.y;

    const float* wt = w + (size_t)task * IN_DIM * OUT_DIM;

    // ---- Stage weights into LDS in WMMA fragment order (coalesced-ish reads) ----
    #pragma unroll
    for (int i = 0; i < (K_STEPS * N_TILES * 32) / 256; ++i) {
        int f    = i * 256 + tid;
        int fl   = f & 31;
        int nt   = (f >> 5) & (N_TILES - 1);
        int ks   = f >> 8;
        int h    = fl >> 4;
        int n    = nt * 16 + (fl & 15);
        int k    = ks * 4 + 2 * h;
        v2f v;
        v.x = wt[(size_t)k * OUT_DIM + n];
        v.y = wt[(size_t)(k + 1) * OUT_DIM + n];
        wfrag[f] = v;
    }
    __syncthreads();

    const int lane16 = lane & 15;
    const int half   = lane >> 4;
    const int m_base = blockIdx.x * ROWS_PER_BLOCK + wave * 16;
    const int row    = m_base + lane16;

    // A fragment source: lane reads in[row][4ks + 2*half .. +1] each k-step.
    const float* arow = in + ((size_t)task * BATCH + row) * IN_DIM + 2 * half;

    // Accumulators, bias folded in: C/D element (vgpr v, lane) has column
    // nt*16 + lane16 for every v, so splat bias across the 8 components.
    v8f acc[N_TILES];
    #pragma unroll
    for (int nt = 0; nt < N_TILES; ++nt) {
        float bv = b[task * OUT_DIM + nt * 16 + lane16];
        v8f a0;
        #pragma unroll
        for (int c = 0; c < 8; ++c) a0[c] = bv;
        acc[nt] = a0;
    }

    // ---- Main K loop: 32 steps of K=4 ----
    #pragma unroll 4
    for (int ks = 0; ks < K_STEPS; ++ks) {
        v2f a = *(const v2f*)(arow + ks * 4);   // global_load_b64, 8B/lane
        #pragma unroll
        for (int nt = 0; nt < N_TILES; ++nt) {
            v2f bf = wfrag[(ks * N_TILES + nt) * 32 + lane];  // ds_load_b64
            acc[nt] = __builtin_amdgcn_wmma_f32_16x16x4_f32(
                /*neg_a=*/false, a, /*neg_b=*/false, bf,
                /*c_mod=*/(short)0, acc[nt],
                /*reuse_a=*/false, /*reuse_b=*/false);
        }
    }

    // ---- Epilogue: C/D layout -> global. vgpr v holds row m_base+v (+8 for hi half),
    // column nt*16 + lane16. ----
    float* obase = out + ((size_t)task * BATCH + m_base) * OUT_DIM;
    #pragma unroll
    for (int nt = 0; nt < N_TILES; ++nt) {
        #pragma unroll
        for (int v = 0; v < 8; ++v) {
            int r = v + 8 * half;
            obase[(size_t)r * OUT_DIM + nt * 16 + lane16] = acc[nt][v];
        }
    }
}

extern "C" void kernel_launch(void* const* d_in, const int* in_sizes, int n_in,
                              void* d_out, int out_size, void* d_ws, size_t ws_size,
                              hipStream_t stream) {
    const float* in = (const float*)d_in[0];   // (8, 32768, 128) f32
    const float* w  = (const float*)d_in[1];   // (8, 128, 128)  f32
    const float* b  = (const float*)d_in[2];   // (8, 128)       f32
    float* out      = (float*)d_out;           // (8, 32768, 128) f32

    dim3 grid(BATCH / ROWS_PER_BLOCK, TASKS);  // (256, 8)
    smd_wmma_f32_kernel<<<grid, 256, 0, stream>>>(in, w, b, out);
}